// SphericalExpansion_10024453669635
// MI455X (gfx1250) — compile-verified
//
#include <hip/hip_runtime.h>

#define N_MAX 8
#define NSH 16
#define N_SPECIES 4
#define N_PSEUDO 4
#define NFEAT 128      // NSH * N_MAX
#define C_STRIDE 512   // N_SPECIES*NFEAT == NSH*N_PSEUDO*N_MAX
#define RCUT 4.0f

typedef float v2f __attribute__((ext_vector_type(2)));
typedef float v8f __attribute__((ext_vector_type(8)));

__global__ __launch_bounds__(256) void zero_ws_kernel(float4* __restrict__ p, int n4) {
    int i = blockIdx.x * blockDim.x + threadIdx.x;
    if (i < n4) p[i] = make_float4(0.f, 0.f, 0.f, 0.f);
}

// Phase 1: per-edge sh[16] x rb[8] outer product, atomically accumulated into
// Acc[c][species][m][n] (L2-resident, 33.5MB).
__global__ __launch_bounds__(256) void edge_kernel(
    const float* __restrict__ vectors,
    const int*   __restrict__ centers,
    const int*   __restrict__ species,
    float*       __restrict__ Acc,
    int E)
{
    int e = blockIdx.x * blockDim.x + threadIdx.x;
    if (e >= E) return;

    float vx = vectors[3*e+0], vy = vectors[3*e+1], vz = vectors[3*e+2];
    float r2 = vx*vx + vy*vy + vz*vz;
    float r  = sqrtf(r2);
    float rs = fmaxf(r, 1e-9f);
    float inv = __frcp_rn(rs);
    float x = vx*inv, y = vy*inv, z = vz*inv;

    // ---- radial basis: rb[n] = sin((n+1)*theta)/((n+1)*theta) * fcut, theta = pi*r/RCUT
    const float theta = r * (3.14159265358979323846f / RCUT);
    float s1, c1;
    __sincosf(theta, &s1, &c1);
    float fcut = (r < RCUT) ? 0.5f * (c1 + 1.0f) : 0.0f;
    float rb[N_MAX];
    float sn_prev = 0.0f;   // sin(0)
    float sn      = s1;     // sin(theta)
    float twoc    = 2.0f * c1;
    #pragma unroll
    for (int n = 1; n <= N_MAX; ++n) {
        float denom = (float)n * theta;
        float sinc  = (denom > 1e-6f) ? (sn / denom) : 1.0f;
        rb[n-1] = sinc * fcut;
        float snext = twoc * sn - sn_prev;
        sn_prev = sn; sn = snext;
    }

    // ---- real spherical harmonics, l = 0..3 (16 values)
    const float K0  = 0.28209479177387814f;   // 0.5*sqrt(1/pi)
    const float K1  = 0.4886025119029199f;    // sqrt(3/(4pi))
    const float K2a = 1.0925484305920792f;    // 0.5*sqrt(15/pi)
    const float K2b = 0.31539156525252005f;   // 0.25*sqrt(5/pi)
    const float K2c = 0.5462742152960396f;    // 0.25*sqrt(15/pi)
    const float K3a = 0.5900435899266435f;    // 0.25*sqrt(35/(2pi))
    const float K3b = 2.890611442640554f;     // 0.5*sqrt(105/pi)
    const float K3c = 0.4570457994644658f;    // 0.25*sqrt(21/(2pi))
    const float K3d = 0.3731763325901154f;    // 0.25*sqrt(7/pi)
    const float K3e = 1.445305721320277f;     // 0.25*sqrt(105/pi)

    float xx = x*x, yy = y*y, zz = z*z;
    float sh[NSH];
    sh[0]  = K0;
    sh[1]  = K1 * y;
    sh[2]  = K1 * z;
    sh[3]  = K1 * x;
    sh[4]  = K2a * x * y;
    sh[5]  = K2a * y * z;
    sh[6]  = K2b * (3.0f*zz - 1.0f);
    sh[7]  = K2a * x * z;
    sh[8]  = K2c * (xx - yy);
    sh[9]  = K3a * y * (3.0f*xx - yy);
    sh[10] = K3b * x * y * z;
    sh[11] = K3c * y * (5.0f*zz - 1.0f);
    sh[12] = K3d * z * (5.0f*zz - 3.0f);
    sh[13] = K3c * x * (5.0f*zz - 1.0f);
    sh[14] = K3e * z * (xx - yy);
    sh[15] = K3a * x * (xx - 3.0f*yy);

    int c = centers[e];
    int s = species[e];
    float* base = Acc + ((size_t)c * C_STRIDE + (size_t)s * NFEAT);
    #pragma unroll
    for (int m = 0; m < NSH; ++m) {
        #pragma unroll
        for (int n = 0; n < N_MAX; ++n) {
            atomicAdd(base + m*N_MAX + n, sh[m] * rb[n]);
        }
    }
}

// Phase 2: out[c, m, p, n] = sum_s W[p, s] * Acc[c, s, m, n]
// GEMM with K=4 -> V_WMMA_F32_16X16X4_F32.
//   A (16x4): rows = 16 "sites" (site = m*8+n within a group), cols = species
//   B (4x16): rows = species, cols = pseudo p (cols 4..15 zero-padded)
//   D (16x16): D[site, p]
// One wave per center; 8 WMMAs cover the 128 sites.
__global__ __launch_bounds__(256) void contract_kernel(
    const float* __restrict__ Acc,
    const float* __restrict__ W,        // W[p][s], row-major (4,4)
    float*       __restrict__ out,
    int C)
{
    int lane = threadIdx.x & 31;
    int wave = (blockIdx.x * blockDim.x + threadIdx.x) >> 5;
    if (wave >= C) return;              // grid exact-fits: never diverges

    int ncol  = lane & 15;              // B: column N; D: column p
    int khalf = lane >> 4;              // K pair selector (0 -> K{0,1}, 1 -> K{2,3})

    // B[k][n] = W[n*4 + k] for n<4, else 0
    v2f b;
    b.x = (ncol < N_PSEUDO) ? W[ncol*N_SPECIES + 2*khalf + 0] : 0.0f;
    b.y = (ncol < N_PSEUDO) ? W[ncol*N_SPECIES + 2*khalf + 1] : 0.0f;

    const float* Ab = Acc + (size_t)wave * C_STRIDE;
    float*       ob = out + (size_t)wave * C_STRIDE;
    int p = ncol;

    #pragma unroll
    for (int g = 0; g < 8; ++g) {
        int site = g*16 + (lane & 15);  // A row index M
        v2f a;
        a.x = Ab[(2*khalf + 0)*NFEAT + site];
        a.y = Ab[(2*khalf + 1)*NFEAT + site];

        v8f acc = {};
        acc = __builtin_amdgcn_wmma_f32_16x16x4_f32(
            /*neg_a=*/false, a, /*neg_b=*/false, b,
            /*c_mod=*/(short)0, acc, /*reuse_a=*/false, /*reuse_b=*/false);

        // D: this lane holds column p = ncol, sites g*16 + 8*khalf + r (r=0..7)
        if (p < N_PSEUDO) {
            // site s = g*16 + 8*khalf + r  ->  m = 2g + khalf, n = r (contiguous)
            float* dst = ob + (2*g + khalf)*32 + p*8;
            *(float4*)(dst + 0) = make_float4(acc[0], acc[1], acc[2], acc[3]);
            *(float4*)(dst + 4) = make_float4(acc[4], acc[5], acc[6], acc[7]);
        }
    }
}

extern "C" void kernel_launch(void* const* d_in, const int* in_sizes, int n_in,
                              void* d_out, int out_size, void* d_ws, size_t ws_size,
                              hipStream_t stream) {
    const float* vectors = (const float*)d_in[0];
    const float* W       = (const float*)d_in[1];
    const int*   centers = (const int*)d_in[2];
    const int*   species = (const int*)d_in[3];
    // d_in[4] is n_centers on device; recover C from out_size instead.
    float* out = (float*)d_out;
    float* Acc = (float*)d_ws;          // needs out_size floats (33.5 MB)

    int E = in_sizes[0] / 3;
    int C = out_size / C_STRIDE;        // 16384

    // zero the accumulator (graph-capture safe)
    {
        int n4 = out_size / 4;
        int blocks = (n4 + 255) / 256;
        zero_ws_kernel<<<blocks, 256, 0, stream>>>((float4*)Acc, n4);
    }
    // phase 1: scatter-accumulate per (center, species)
    {
        int blocks = (E + 255) / 256;
        edge_kernel<<<blocks, 256, 0, stream>>>(vectors, centers, species, Acc, E);
    }
    // phase 2: WMMA contraction with W (one wave per center, 8 waves/block)
    {
        int blocks = (C + 7) / 8;       // C divisible by 8 -> EXEC always full
        contract_kernel<<<blocks, 256, 0, stream>>>(Acc, W, out, C);
    }
}